// CrossPatchModule_83949430768099
// MI455X (gfx1250) — compile-verified
//
#include <hip/hip_runtime.h>

typedef unsigned int u32;
typedef unsigned long long u64;
typedef __attribute__((ext_vector_type(4))) u32   u32x4;
typedef __attribute__((ext_vector_type(8))) int   i32x8;
typedef __attribute__((ext_vector_type(4))) float f32x4;

// Problem constants (from reference): B=8, C=64, H=W=512, PH=PW=8 -> PN=64, kh=kw=64.
#define IMG       512u
#define TILE      64u                 // kh = kw = 64
#define TILE_F4   1024u               // 64*64 floats / 4
#define THREADS   256u

__global__ __launch_bounds__(THREADS)
void crosspatch_tdm_kernel(const float* __restrict__ x,
                           const float* __restrict__ pos,
                           float* __restrict__ out)
{
    __shared__ float tile[TILE * TILE];   // 16 KiB staging tile

    const u32 bid = blockIdx.x;
    const u32 p   = bid & 63u;            // output patch index (i*8+j)
    const u32 c   = (bid >> 6) & 63u;     // channel
    const u32 b   = bid >> 12;            // batch
    const u32 p2  = (p + c) & 63u;        // source patch index

    const u32 bc = b * 64u + c;
    const u64 plane = (u64)bc * (u64)(IMG * IMG);
    const u64 srcOff = plane + (u64)(p2 >> 3) * (TILE * IMG) + (u64)(p2 & 7u) * TILE;
    const u64 dstOff = plane + (u64)(p  >> 3) * (TILE * IMG) + (u64)(p  & 7u) * TILE;

    // one scalar bias per output patch (broadcast over the 64x64 tile); uniform -> SMEM load
    const float posv = pos[c * 64u + p2];

    if (threadIdx.x < 32u) {
        // ---- Build Tensor DMA Descriptor (D#) for a 64x64 fp32 tile ----
        const u64 gaddr   = (u64)(const void*)(x + srcOff);       // byte address of tile start
        const u32 ldsAddr = (u32)(u64)(void*)&tile[0];            // LDS byte offset (low 32 bits of flat addr)

        // Group 0 (4 SGPRs): count=1 | lds_addr | global_addr[56:0] | type=2
        u32x4 g0;
        g0.x = 1u;                                    // count=1 (valid), is_restore=0, gather off
        g0.y = ldsAddr;                               // lds_addr
        g0.z = (u32)(gaddr & 0xFFFFFFFFull);          // global_addr[31:0]
        g0.w = (u32)((gaddr >> 32) & 0x1FFFFFFull)    // global_addr[56:32]
             | (2u << 30);                            // type = 2 ("image")

        // Group 1 (8 SGPRs): data_size=4B, tile 64x64, row stride 512 elems
        const u32 tdim0 = 1u << 24;                   // generous tensor dims: tile never clipped
        const u32 tdim1 = 1u << 24;
        i32x8 g1;
        g1[0] = (int)(2u << 16);                                              // wg_mask=0, data_size=2 (4B)
        g1[1] = (int)((tdim0 & 0xFFFFu) << 16);                               // barrier_addr=0 | tensor_dim0[15:0]
        g1[2] = (int)(((tdim0 >> 16) & 0xFFFFu) | ((tdim1 & 0xFFFFu) << 16)); // tensor_dim0[31:16] | tensor_dim1[15:0]
        g1[3] = (int)(((tdim1 >> 16) & 0xFFFFu) | (TILE << 16));              // tensor_dim1[31:16] | tile_dim0=64
        g1[4] = (int)TILE;                                                    // tile_dim1=64, tile_dim2=0 (2D)
        g1[5] = (int)IMG;                                                     // tensor_dim0_stride = 512 elements
        g1[6] = 0;                                                            // stride0 hi | stride1 lo (unused, 2D)
        g1[7] = 0;                                                            // stride1 hi (unused)

        // Issue the DMA: tensor_load_to_lds VADDR0(group0, 4 SGPRs), VADDR1(group1, 8 SGPRs)
        asm volatile("tensor_load_to_lds %0, %1"
                     :
                     : "s"(g0), "s"(g1)
                     : "memory");

        // Wait for the TDM transfer to land in LDS (TENSORcnt == 0)
        __builtin_amdgcn_s_wait_tensorcnt(0);
    }
    __syncthreads();

    // ---- Drain LDS -> global with bias add ----
    // 128-bit LDS reads, bias add, 128-bit fully-coalesced NON-TEMPORAL global
    // stores (stream-once output; NT avoids rinsing the 192MB L2).
    const f32x4* __restrict__ lt = (const f32x4*)tile;
    float* __restrict__ dst = out + dstOff;

#pragma unroll
    for (u32 k = 0; k < TILE_F4 / THREADS; ++k) {
        const u32 e  = threadIdx.x + k * THREADS;   // float4 index within tile [0,1024)
        const u32 a  = e >> 4;                      // row within tile   [0,64)
        const u32 d4 = e & 15u;                     // float4 col        [0,16)
        f32x4 v = lt[e];
        v += posv;                                  // vector + scalar broadcast add
        __builtin_nontemporal_store(v, (f32x4*)(dst + (u64)a * IMG + d4 * 4u));
    }
}

extern "C" void kernel_launch(void* const* d_in, const int* in_sizes, int n_in,
                              void* d_out, int out_size, void* d_ws, size_t ws_size,
                              hipStream_t stream)
{
    const float* x   = (const float*)d_in[0];   // [8,64,512,512] fp32
    const float* pos = (const float*)d_in[1];   // [64,64] fp32 (abs_pos[0,0,:,:,0,0])
    float*       out = (float*)d_out;           // [8,64,512,512] fp32

    (void)in_sizes; (void)n_in; (void)out_size; (void)d_ws; (void)ws_size;

    const u32 nblocks = 8u * 64u * 64u;         // one block per (b, c, patch)
    crosspatch_tdm_kernel<<<dim3(nblocks), dim3(THREADS), 0, stream>>>(x, pos, out);
}